// GATNet_80633716015692
// MI455X (gfx1250) — compile-verified
//
#include <hip/hip_runtime.h>
#include <hip/hip_bf16.h>

// ---------------------------------------------------------------------------
// GAT (2-layer) for MI455X / gfx1250.
//  - GEMMs: V_WMMA_F32_16X16X4_F32 (exact fp32 matrix pipe)
//  - Operand staging: GLOBAL_LOAD_ASYNC_TO_LDS_B128 + s_wait_asynccnt
//    (guarded; falls back to float4 + ds_store if builtins absent)
// ---------------------------------------------------------------------------

typedef float v2f __attribute__((ext_vector_type(2)));
typedef float v8f __attribute__((ext_vector_type(8)));
typedef int   v4i __attribute__((vector_size(16)));   // matches builtin param

static constexpr int   NN    = 20000;    // nodes
static constexpr int   OC    = 128;      // per-head output channels
static constexpr int   H1    = 4;        // heads layer 1
static constexpr int   NE    = 320000;   // real edges
static constexpr int   NT    = NE + NN;  // + self loops
static constexpr float SLOPE = 0.2f;

static constexpr int GK  = 512;          // shared K for both GEMMs
static constexpr int LDA = GK + 4;       // A LDS stride: 4*idx banks, conflict-free
static constexpr int KC  = 32;           // B K-chunk
static constexpr int LDB = 136;          // B LDS stride: half-wave K offsets hit
                                         // disjoint bank ranges (2*136 % 64 == 16)

// -------------------- async global->LDS staging (gfx1250) ------------------
#if __has_builtin(__builtin_amdgcn_global_load_async_to_lds_b128) && \
    __has_builtin(__builtin_amdgcn_s_wait_asynccnt)
#define USE_ASYNC_LDS 1
#else
#define USE_ASYNC_LDS 0
#endif

#if USE_ASYNC_LDS
#define AS_GLOBAL __attribute__((address_space(1)))
#define AS_LDS    __attribute__((address_space(3)))
__device__ __forceinline__ void stage16(const float* __restrict__ g,
                                        float* __restrict__ l) {
  // 16-byte per-lane copy, tracked with ASYNCcnt (no VGPR round trip).
  __builtin_amdgcn_global_load_async_to_lds_b128(
      (AS_GLOBAL v4i*)g, (AS_LDS v4i*)l, 0, 0);
}
__device__ __forceinline__ void stage_wait() {
  __builtin_amdgcn_s_wait_asynccnt(0);
}
#else
__device__ __forceinline__ void stage16(const float* __restrict__ g,
                                        float* __restrict__ l) {
  *reinterpret_cast<float4*>(l) = *reinterpret_cast<const float4*>(g);
}
__device__ __forceinline__ void stage_wait() {}
#endif

// -------------------- fp32 WMMA GEMM: C[M,N] = A[M,512] * B[512,N] ---------
// block = 256 threads (8 waves) -> 16 x 128 strip of C.
// gridDim.y = M/16, gridDim.x = N/128 (N in {512,128}).
__global__ __launch_bounds__(256)
void wmma_gemm_k512(const float* __restrict__ A, const float* __restrict__ B,
                    float* __restrict__ C, int N) {
  __shared__ float As[16 * LDA];   // 33,024 B
  __shared__ float Bs[KC * LDB];   // 17,408 B

  const int tid    = threadIdx.x;
  const int tile_m = blockIdx.y * 16;
  const int bn0    = blockIdx.x * 128;

  // ---- stage A tile 16x512: 2048 16B-granules / 256 threads = 8 each ----
#pragma unroll
  for (int i = 0; i < 8; ++i) {
    const int f4 = tid + i * 256;
    const int r  = f4 >> 7;                 // 128 granules per row
    const int c4 = f4 & 127;
    stage16(A + (size_t)(tile_m + r) * GK + c4 * 4, &As[r * LDA + c4 * 4]);
  }

  const int wave = tid >> 5;
  const int lane = tid & 31;
  const int half = lane >> 4;               // selects K pair {k,k+1} vs {k+2,k+3}
  const int idx  = lane & 15;               // M (A) / N (B,C) index
  const int ncol = wave * 16 + idx;         // column within 128-wide strip

  v8f acc = {0.f, 0.f, 0.f, 0.f, 0.f, 0.f, 0.f, 0.f};

  for (int kb = 0; kb < GK; kb += KC) {
    // ---- stage B chunk KCx128: 1024 granules / 256 threads = 4 each ----
#pragma unroll
    for (int i = 0; i < 4; ++i) {
      const int f4 = tid + i * 256;
      const int r  = f4 >> 5;               // 32 granules per row
      const int c4 = f4 & 31;
      stage16(B + (size_t)(kb + r) * N + bn0 + c4 * 4, &Bs[r * LDB + c4 * 4]);
    }
    stage_wait();          // this wave's async copies (A on first iter too)
    __syncthreads();       // everyone's copies visible

#pragma unroll
    for (int k = 0; k < KC; k += 4) {
      const int kk = k + (half << 1);
      // A 16x4 (ISA 7.12.2): lane idx = M, VGPR0/1 = K kk,kk+1
      v2f a = *reinterpret_cast<const v2f*>(&As[idx * LDA + kb + kk]);
      // B 4x16 (mirrors A): lane idx = N, VGPR0/1 = K kk,kk+1
      v2f b;
      b.x = Bs[kk * LDB + ncol];
      b.y = Bs[(kk + 1) * LDB + ncol];
      acc = __builtin_amdgcn_wmma_f32_16x16x4_f32(
          /*neg_a=*/false, a, /*neg_b=*/false, b,
          /*c_mod=*/(short)0, acc, /*reuse_a=*/false, /*reuse_b=*/false);
    }
    __syncthreads();       // chunk consumed before next overwrite
  }

  // C/D layout: VGPR v -> M = v + 8*half, N = idx.
#pragma unroll
  for (int v = 0; v < 8; ++v) {
    const int row = tile_m + v + (half << 3);
    C[(size_t)row * N + bn0 + wave * 16 + idx] = acc[v];
  }
}

// -------------------- attention coefficients al = <h, a> -------------------
// one wave per (node, head); 32 lanes x float4 = 128 channels.
__global__ __launch_bounds__(256)
void row_dot(const float* __restrict__ feat, const float* __restrict__ a_src,
             const float* __restrict__ a_dst, float* __restrict__ out_src,
             float* __restrict__ out_dst, int heads) {
  const int wid  = (int)((blockIdx.x * (unsigned)blockDim.x + threadIdx.x) >> 5);
  const int lane = threadIdx.x & 31;
  if (wid >= NN * heads) return;
  const int node = wid / heads;
  const int head = wid - node * heads;

  const size_t fb = (size_t)node * heads * OC + (size_t)head * OC + lane * 4;
  const float4 h = *reinterpret_cast<const float4*>(feat + fb);
  const float4 s = *reinterpret_cast<const float4*>(a_src + head * OC + lane * 4);
  const float4 d = *reinterpret_cast<const float4*>(a_dst + head * OC + lane * 4);

  float ps = h.x * s.x + h.y * s.y + h.z * s.z + h.w * s.w;
  float pd = h.x * d.x + h.y * d.y + h.z * d.z + h.w * d.w;
#pragma unroll
  for (int off = 16; off > 0; off >>= 1) {
    ps += __shfl_xor(ps, off, 32);
    pd += __shfl_xor(pd, off, 32);
  }
  if (lane == 0) { out_src[wid] = ps; out_dst[wid] = pd; }
}

// -------------------- edge helpers -----------------------------------------
__device__ __forceinline__ void edge_sd(const int* __restrict__ ei, int e,
                                        int& s, int& d) {
  if (e < NE) { s = ei[e]; d = ei[NE + e]; }
  else        { s = e - NE; d = s; }            // appended self loops
}
__device__ __forceinline__ int edge_d(const int* __restrict__ ei, int e) {
  return (e < NE) ? ei[NE + e] : (e - NE);
}

// monotone float<->uint order mapping (for atomicMax on floats)
__device__ __forceinline__ unsigned enc_f(float f) {
  unsigned u = __float_as_uint(f);
  return (u & 0x80000000u) ? ~u : (u | 0x80000000u);
}
__device__ __forceinline__ float dec_f(unsigned u) {
  return (u & 0x80000000u) ? __uint_as_float(u ^ 0x80000000u)
                           : __uint_as_float(~u);
}
static constexpr unsigned ENC_NEG_INF = 0x007FFFFFu;   // enc(-inf)

// pass 1: leaky-relu logits + segment max over dst
__global__ void edge_logits_max(const int* __restrict__ ei,
                                const float* __restrict__ alsrc,
                                const float* __restrict__ aldst,
                                float* __restrict__ ev,
                                unsigned* __restrict__ mx, int heads) {
  const int e = blockIdx.x * blockDim.x + threadIdx.x;
  if (e >= NT) return;
  int s, d; edge_sd(ei, e, s, d);
  for (int h = 0; h < heads; ++h) {
    float v = alsrc[s * heads + h] + aldst[d * heads + h];
    v = (v > 0.f) ? v : SLOPE * v;
    ev[(size_t)e * heads + h] = v;
    atomicMax(&mx[d * heads + h], enc_f(v));
  }
}

// pass 2: exp(v - max) + segment sum over dst (overwrites ev with numerator)
__global__ void edge_exp_sum(const int* __restrict__ ei,
                             float* __restrict__ ev,
                             const unsigned* __restrict__ mx,
                             float* __restrict__ den, int heads) {
  const int e = blockIdx.x * blockDim.x + threadIdx.x;
  if (e >= NT) return;
  const int d = edge_d(ei, e);
  for (int h = 0; h < heads; ++h) {
    const float m = dec_f(mx[d * heads + h]);
    const float p = expf(ev[(size_t)e * heads + h] - m);
    ev[(size_t)e * heads + h] = p;
    atomicAdd(&den[d * heads + h], p);
  }
}

// pass 3: out[dst] += alpha * feat[src]   (one block per edge)
__global__ void edge_aggregate(const int* __restrict__ ei,
                               const float* __restrict__ feat,
                               const float* __restrict__ ev,
                               const float* __restrict__ den,
                               float* __restrict__ out, int heads, int hc) {
  const int e = blockIdx.x;
  int s, d; edge_sd(ei, e, s, d);
  for (int c = threadIdx.x; c < hc; c += blockDim.x) {
    const int h = c >> 7;                          // OC == 128 per head
    const float alpha =
        ev[(size_t)e * heads + h] / (den[d * heads + h] + 1e-16f);
    atomicAdd(&out[(size_t)d * hc + c], alpha * feat[(size_t)s * hc + c]);
  }
}

// -------------------- small elementwise kernels ----------------------------
__global__ void fill_bias(float* __restrict__ out, const float* __restrict__ b,
                          long total, int cmask) {
  const long i = blockIdx.x * (long)blockDim.x + threadIdx.x;
  if (i < total) out[i] = b[i & cmask];            // c is a power of two
}

__global__ void init_softmax(unsigned* __restrict__ mx, float* __restrict__ den,
                             int cnt) {
  const int i = blockIdx.x * blockDim.x + threadIdx.x;
  if (i < cnt) { mx[i] = ENC_NEG_INF; den[i] = 0.f; }
}

__global__ void elu_inplace(float* __restrict__ x, long total) {
  const long i = blockIdx.x * (long)blockDim.x + threadIdx.x;
  if (i < total) {
    const float v = x[i];
    x[i] = (v > 0.f) ? v : expm1f(v);
  }
}

// ---------------------------------------------------------------------------
extern "C" void kernel_launch(void* const* d_in, const int* in_sizes, int n_in,
                              void* d_out, int out_size, void* d_ws, size_t ws_size,
                              hipStream_t stream) {
  const float* x      = (const float*)d_in[0];
  const int*   edge   = (const int*)  d_in[1];   // [2, NE] int32
  const float* W1     = (const float*)d_in[2];   // [512, 512]
  const float* a_src1 = (const float*)d_in[3];   // [4, 128]
  const float* a_dst1 = (const float*)d_in[4];
  const float* b1     = (const float*)d_in[5];   // [512]
  const float* W2     = (const float*)d_in[6];   // [512, 128]
  const float* a_src2 = (const float*)d_in[7];   // [1, 128]
  const float* a_dst2 = (const float*)d_in[8];
  const float* b2     = (const float*)d_in[9];   // [128]
  float*       out    = (float*)d_out;           // [20000, 128]

  // ---- workspace arena (floats) ----
  float* wsf = (float*)d_ws;
  float*    h1    = wsf;                         // 10,240,000  [N,512] lin feats
  float*    out1  = wsf + 10240000;              // 10,240,000  [N,512] aggregated
  float*    alsrc = wsf + 20480000;              //     80,000
  float*    aldst = wsf + 20560000;              //     80,000
  unsigned* mx    = (unsigned*)(wsf + 20640000); //     80,000
  float*    den   = wsf + 20720000;              //     80,000
  float*    ev    = wsf + 20800000;              //  1,360,000  [NT, H]
  // layer-2 reuses h1 slot for its linear features and the same softmax bufs.

  const long n1 = (long)NN * (H1 * OC);          // 10.24M
  const long n2 = (long)NN * OC;                 // 2.56M
  const int  egrid = (NT + 255) / 256;

  // ================= layer 1 (H=4, concat) =================
  fill_bias   <<<(n1 + 255) / 256, 256, 0, stream>>>(out1, b1, n1, H1 * OC - 1);
  init_softmax<<<(NN * H1 + 255) / 256, 256, 0, stream>>>(mx, den, NN * H1);

  wmma_gemm_k512<<<dim3((H1 * OC) / 128, NN / 16), 256, 0, stream>>>(x, W1, h1, H1 * OC);

  row_dot<<<(NN * H1 + 7) / 8, 256, 0, stream>>>(h1, a_src1, a_dst1, alsrc, aldst, H1);

  edge_logits_max<<<egrid, 256, 0, stream>>>(edge, alsrc, aldst, ev, mx, H1);
  edge_exp_sum   <<<egrid, 256, 0, stream>>>(edge, ev, mx, den, H1);
  edge_aggregate <<<NT, 256, 0, stream>>>(edge, h1, ev, den, out1, H1, H1 * OC);

  elu_inplace<<<(n1 + 255) / 256, 256, 0, stream>>>(out1, n1);

  // ================= layer 2 (H=1, mean == identity) =================
  float* h2 = h1;                                // reuse: [N,128] linear feats
  fill_bias   <<<(n2 + 255) / 256, 256, 0, stream>>>(out, b2, n2, OC - 1);
  init_softmax<<<(NN + 255) / 256, 256, 0, stream>>>(mx, den, NN);

  wmma_gemm_k512<<<dim3(OC / 128, NN / 16), 256, 0, stream>>>(out1, W2, h2, OC);

  row_dot<<<(NN + 7) / 8, 256, 0, stream>>>(h2, a_src2, a_dst2, alsrc, aldst, 1);

  edge_logits_max<<<egrid, 256, 0, stream>>>(edge, alsrc, aldst, ev, mx, 1);
  edge_exp_sum   <<<egrid, 256, 0, stream>>>(edge, ev, mx, den, 1);
  edge_aggregate <<<NT, 128, 0, stream>>>(edge, h2, ev, den, out, 1, OC);

  (void)in_sizes; (void)n_in; (void)out_size; (void)ws_size;
}